// HexaToParallelogram_52304111731362
// MI455X (gfx1250) — compile-verified
//
#include <hip/hip_runtime.h>

// HexaToParallelogram gather for MI455X (gfx1250) — round 2.
//
// Pure memory permutation: 0 FLOPs, ~314 MB HBM traffic -> ~13.5 us floor at
// 23.3 TB/s. WMMA is inapplicable; the chip-relevant paths are the memory
// system: nontemporal load/store hints, global_prefetch_b8, and LDS (ds_load)
// to amortize the constant 37x37 index map.
//
// Decomposition: one block = 4 input rows = 5476 output floats = 1369 aligned
// float4 stores (4*1369*4 B = 21904 B, multiple of 16 -> every block's vec4
// region is 16B-aligned). 32768 rows / 4 = 8192 blocks exactly. This removes
// the per-element /1369 and /37 divisions: the hex index table is built once
// per block into LDS (u16, 2.7 KB), and row-within-chunk comes from three
// compares. Hot loop: ds_load_u16 -> predicated global_load_b32 (NT) ->
// global_store_b128 (NT). ~8 VALU/elem vs ~20+ in the divide-per-element
// version, keeping VALU well under the rate needed to saturate HBM.

typedef float v4f __attribute__((ext_vector_type(4)));

#define HEX_R    18
#define NPIX     1027u   // valid hex pixels; idx >= NPIX -> padding (0.0f)
#define INSTRIDE 1039u   // input last-dim stride (only first 1027 used)
#define CELLS    1369u   // 37*37 output cells per row
#define RPB      4u      // rows per block
#define NTHREADS 256u    // 8 wave32

// Closed-form replica of _build_lookup(): cell = qi*37 + ri, q = qi-18,
// r = ri-18. Valid iff |q+r| <= 18. idx = offset(q) + (r - r_lo(q)) with
//   q <= 0: r_lo = -18-q, offset = m*(R+1) + m(m-1)/2,  m = q+R
//   q >  0: r_lo = -18,   offset = 495 + 37q - q(q-1)/2
// Spot-checked: offset(0)=495, idx(q=18,r=0)=1026, offset(19)=1027.
__device__ __forceinline__ unsigned hex_index(unsigned cell) {
    unsigned qi = cell / 37u;
    unsigned ri = cell - qi * 37u;
    int q = (int)qi - HEX_R;
    int r = (int)ri - HEX_R;
    int s = q + r;
    if ((unsigned)(s + HEX_R) > 36u) return NPIX;   // corner -> padding slot
    if (q <= 0) {
        int m = q + HEX_R;
        int off = m * (HEX_R + 1) + ((m * (m - 1)) >> 1);
        return (unsigned)(off + r + HEX_R + q);
    } else {
        int off = 495 + q * 37 - ((q * (q - 1)) >> 1);
        return (unsigned)(off + r + HEX_R);
    }
}

__global__ __launch_bounds__(NTHREADS)
void HexaToParallelogram_52304111731362_kernel(const float* __restrict__ in,
                                               float* __restrict__ out,
                                               unsigned n_rows) {
    __shared__ unsigned short idxtab[CELLS + 8];   // +pad, 2.7 KB of 320 KB

    const unsigned tid  = threadIdx.x;
    const unsigned blk  = blockIdx.x;
    const unsigned row0 = blk * RPB;
    if (row0 >= n_rows) return;

    const float* __restrict__ inb = in + (size_t)row0 * INSTRIDE;

    // Prefetch this block's entire input chunk (4 rows * 1039 * 4B ~= 16.6 KB)
    // while we build the index table. Emits global_prefetch_b8.
    for (unsigned o = tid * 16u; o < RPB * INSTRIDE; o += NTHREADS * 16u)
        __builtin_prefetch(inb + o, 0, 1);

    // Build the constant 37x37 index map once per block (1369 / 256 ~= 5.4
    // entries per thread). u16 is plenty (max value 1027 sentinel).
    for (unsigned c = tid; c < CELLS; c += NTHREADS)
        idxtab[c] = (unsigned short)hex_index(c);
    __syncthreads();

    v4f* __restrict__ out4 = (v4f*)out + (size_t)blk * CELLS;

    if (row0 + RPB <= n_rows) {
        // Fast path (always taken for the reference shapes): 1369 vec4s.
        for (unsigned j = tid; j < CELLS; j += NTHREADS) {
            const unsigned p0 = j * 4u;
            v4f v;
#pragma unroll
            for (int k = 0; k < 4; ++k) {
                unsigned p    = p0 + (unsigned)k;
                unsigned rc   = (unsigned)(p >= CELLS) + (unsigned)(p >= 2u * CELLS)
                              + (unsigned)(p >= 3u * CELLS);      // row in chunk
                unsigned cell = p - rc * CELLS;
                unsigned idx  = idxtab[cell];                      // ds_load_u16
                float x = 0.0f;
                if (idx < NPIX)
                    x = __builtin_nontemporal_load(inb + rc * INSTRIDE + idx);
                v[k] = x;
            }
            __builtin_nontemporal_store(v, out4 + j);              // b128, 16B-aligned
        }
    } else {
        // Ragged tail (cold; only if n_rows % 4 != 0): scalar guarded stores.
        const unsigned nval = (n_rows - row0) * CELLS;             // valid elems
        float* __restrict__ outs = (float*)out4;
        for (unsigned j = tid; j < CELLS; j += NTHREADS) {
            const unsigned p0 = j * 4u;
            for (int k = 0; k < 4; ++k) {
                unsigned p = p0 + (unsigned)k;
                if (p >= nval) continue;
                unsigned rc   = (unsigned)(p >= CELLS) + (unsigned)(p >= 2u * CELLS)
                              + (unsigned)(p >= 3u * CELLS);
                unsigned cell = p - rc * CELLS;
                unsigned idx  = idxtab[cell];
                outs[p] = (idx < NPIX) ? inb[rc * INSTRIDE + idx] : 0.0f;
            }
        }
    }
}

extern "C" void kernel_launch(void* const* d_in, const int* in_sizes, int n_in,
                              void* d_out, int out_size, void* d_ws, size_t ws_size,
                              hipStream_t stream) {
    const float* in  = (const float*)d_in[0];
    float*       out = (float*)d_out;
    unsigned n_rows = (unsigned)in_sizes[0] / INSTRIDE;   // 64*512 = 32768
    unsigned blocks = (n_rows + RPB - 1u) / RPB;          // 8192
    HexaToParallelogram_52304111731362_kernel<<<blocks, NTHREADS, 0, stream>>>(
        in, out, n_rows);
}